// GNN_BAMultiShapesDataset_64046552317956
// MI455X (gfx1250) — compile-verified
//
#include <hip/hip_runtime.h>
#include <hip/hip_bf16.h>
#include <math.h>

#define HID 64

typedef float v2f __attribute__((ext_vector_type(2)));
typedef float v8f __attribute__((ext_vector_type(8)));

// ---------------- utility ----------------
__global__ void zero_f32(float* __restrict__ p, size_t n) {
  size_t i = (size_t)blockIdx.x * blockDim.x + threadIdx.x;
  size_t stride = (size_t)gridDim.x * blockDim.x;
  for (; i < n; i += stride) p[i] = 0.0f;
}

// ---------------- gcn norm ----------------
__global__ void deg_kernel(const int* __restrict__ col, float* __restrict__ deg, int E) {
  int e = blockIdx.x * blockDim.x + threadIdx.x;
  if (e < E) unsafeAtomicAdd(&deg[col[e]], 1.0f);
}

__global__ void dinv_kernel(float* __restrict__ deg, int n) {
  int i = blockIdx.x * blockDim.x + threadIdx.x;
  if (i < n) {
    float d = deg[i];
    deg[i] = (d > 0.0f) ? rsqrtf(d) : 0.0f;
  }
}

__global__ void norm_kernel(const int* __restrict__ row, const int* __restrict__ col,
                            const float* __restrict__ dinv, float* __restrict__ norm, int E) {
  int e = blockIdx.x * blockDim.x + threadIdx.x;
  if (e < E) norm[e] = dinv[row[e]] * dinv[col[e]];
}

// ---------------- layer-1 transform: out[N,64] = x[N,10] @ W1[10,64] ----------------
__global__ __launch_bounds__(256)
void xform_in(const float* __restrict__ x, const float* __restrict__ W1,
              float* __restrict__ out, int n_nodes) {
  __shared__ float sW[10 * HID];
  for (int i = threadIdx.x; i < 10 * HID; i += 256) sW[i] = W1[i];
  __syncthreads();
  int node = blockIdx.x * 4 + (threadIdx.x >> 6);
  int f    = threadIdx.x & 63;
  if (node >= n_nodes) return;
  const float* xr = x + (size_t)node * 10;
  float s = 0.0f;
#pragma unroll
  for (int k = 0; k < 10; ++k) s += xr[k] * sW[k * HID + f];
  out[(size_t)node * HID + f] = s;
}

// ---------------- WMMA GEMM: C[N,64] = A[N,64] @ W[64,64] (all f32) ----------------
// One wave per 16x16 output tile; K=64 consumed in 16 steps of V_WMMA_F32_16X16X4_F32.
// A-frag layout (ISA 7.12.2): lanes 0-15 hold M=lane, {K=k,k+1}; lanes 16-31 {K=k+2,k+3}.
// B-frag: lanes 0-15 hold N=lane rows {K=k,k+1}; lanes 16-31 rows {K=k+2,k+3}.
// C/D: VGPR r -> M = r + 8*(lane>=16), N = lane&15.
__global__ __launch_bounds__(256)
void gemm64_wmma(const float* __restrict__ A, const float* __restrict__ W,
                 float* __restrict__ C, int n_nodes) {
  int wave = threadIdx.x >> 5;       // 0..7
  int lane = threadIdx.x & 31;
  int nodeTile = blockIdx.x * 2 + (wave >> 2);   // 2 node tiles per block
  int featTile = wave & 3;                       // 4 feature tiles of 16
  int row0 = nodeTile * 16;
  if (row0 >= n_nodes) return;                   // wave-uniform
  int m  = lane & 15;
  int kh = lane >> 4;                            // 0 or 1: selects K pair
  int col0 = featTile * 16;

  const float* Arow = A + (size_t)(row0 + m) * HID;
  v8f acc = {};
#pragma unroll
  for (int k = 0; k < HID; k += 4) {
    float2 av = *reinterpret_cast<const float2*>(Arow + k + 2 * kh);
    v2f a; a.x = av.x; a.y = av.y;
    v2f b;
    b.x = W[(k + 2 * kh + 0) * HID + col0 + m];
    b.y = W[(k + 2 * kh + 1) * HID + col0 + m];
    acc = __builtin_amdgcn_wmma_f32_16x16x4_f32(false, a, false, b, (short)0, acc,
                                                false, false);
  }
#pragma unroll
  for (int r = 0; r < 8; ++r) {
    int mm = r + kh * 8;
    C[(size_t)(row0 + mm) * HID + col0 + m] = acc[r];
  }
}

// ---------------- edge scatter: agg[col] += t[row] * norm ----------------
// 16 threads per edge, each thread moves 4 consecutive features (float4 gather,
// 4 hardware f32 atomics to L2).
__global__ __launch_bounds__(256)
void scatter_kernel(const float* __restrict__ t, const float* __restrict__ norm,
                    const int* __restrict__ row, const int* __restrict__ col,
                    float* __restrict__ agg, int E) {
  size_t tid = (size_t)blockIdx.x * blockDim.x + threadIdx.x;
  size_t e = tid >> 4;
  int q = (int)(tid & 15);
  if (e >= (size_t)E) return;
  int r = row[e], c = col[e];
  float nm = norm[e];
  float4 v = reinterpret_cast<const float4*>(t + (size_t)r * HID)[q];
  float* dst = agg + (size_t)c * HID + q * 4;
  unsafeAtomicAdd(dst + 0, v.x * nm);
  unsafeAtomicAdd(dst + 1, v.y * nm);
  unsafeAtomicAdd(dst + 2, v.z * nm);
  unsafeAtomicAdd(dst + 3, v.w * nm);
}

// ---------------- bias (+ optional relu) in place ----------------
template <bool RELU>
__global__ void bias_act(float* __restrict__ h, const float* __restrict__ b, size_t n) {
  size_t i = (size_t)blockIdx.x * blockDim.x + threadIdx.x;
  if (i >= n) return;
  float v = h[i] + b[i & (HID - 1)];
  if (RELU) v = fmaxf(v, 0.0f);
  h[i] = v;
}

// ---------------- global add pool ----------------
__global__ void pool_kernel(const float* __restrict__ h, const int* __restrict__ batch,
                            float* __restrict__ gemb, size_t n) {
  size_t i = (size_t)blockIdx.x * blockDim.x + threadIdx.x;
  if (i >= n) return;
  int g = batch[i >> 6];
  unsafeAtomicAdd(&gemb[(size_t)g * HID + (i & (HID - 1))], h[i]);
}

// ---------------- head: logits = gemb @ Wl + bl ; softmax(2) ----------------
__global__ void head_kernel(const float* __restrict__ gemb, const float* __restrict__ Wl,
                            const float* __restrict__ bl, float* __restrict__ out,
                            int n_graphs) {
  int g = blockIdx.x * blockDim.x + threadIdx.x;
  if (g >= n_graphs) return;
  const float* e = gemb + (size_t)g * HID;
  float l0 = bl[0], l1 = bl[1];
#pragma unroll
  for (int f = 0; f < HID; ++f) {
    float v = e[f];
    l0 += v * Wl[f * 2 + 0];
    l1 += v * Wl[f * 2 + 1];
  }
  float mx = fmaxf(l0, l1);
  float e0 = __expf(l0 - mx), e1 = __expf(l1 - mx);
  float inv = 1.0f / (e0 + e1);
  out[(size_t)g * 2 + 0] = e0 * inv;
  out[(size_t)g * 2 + 1] = e1 * inv;
}

extern "C" void kernel_launch(void* const* d_in, const int* in_sizes, int n_in,
                              void* d_out, int out_size, void* d_ws, size_t ws_size,
                              hipStream_t stream) {
  const float* x    = (const float*)d_in[0];
  const int*   ei   = (const int*)d_in[1];
  const int*   bat  = (const int*)d_in[2];
  const float* W1   = (const float*)d_in[3];
  const float* b1   = (const float*)d_in[4];
  const float* W2   = (const float*)d_in[5];
  const float* b2   = (const float*)d_in[6];
  const float* W3   = (const float*)d_in[7];
  const float* b3   = (const float*)d_in[8];
  const float* Wl   = (const float*)d_in[9];
  const float* bl   = (const float*)d_in[10];
  float* out = (float*)d_out;

  const int N = in_sizes[0] / 10;       // 100000
  const int E = in_sizes[1] / 2;        // 3200000
  const int NG = out_size / 2;          // 1000
  const int* row = ei;
  const int* col = ei + E;

  // workspace layout (floats)
  float* ws   = (float*)d_ws;
  float* dinv = ws;                              // N
  float* nrm  = dinv + N;                        // E
  float* bufT = nrm + E;                         // N*HID (transform output)
  float* bufH = bufT + (size_t)N * HID;          // N*HID (agg / layer input)
  float* gemb = bufH + (size_t)N * HID;          // NG*HID
  (void)ws_size; (void)n_in;

  const int TB = 256;
  const size_t NH = (size_t)N * HID;
  dim3 blkNH((unsigned)((NH + TB - 1) / TB));
  dim3 blkE((unsigned)((E + TB - 1) / TB));
  dim3 blkScatter((unsigned)(((size_t)E * 16 + TB - 1) / TB));
  dim3 blkGemm((unsigned)(N / 32));              // 2 node tiles of 16 per block

  // ---- gcn norm ----
  zero_f32<<<1024, TB, 0, stream>>>(dinv, (size_t)N);
  deg_kernel<<<blkE, TB, 0, stream>>>(col, dinv, E);
  dinv_kernel<<<(N + TB - 1) / TB, TB, 0, stream>>>(dinv, N);
  norm_kernel<<<blkE, TB, 0, stream>>>(row, col, dinv, nrm, E);

  // ---- layer 1 ----
  xform_in<<<(N + 3) / 4, TB, 0, stream>>>(x, W1, bufT, N);
  zero_f32<<<2048, TB, 0, stream>>>(bufH, NH);
  scatter_kernel<<<blkScatter, TB, 0, stream>>>(bufT, nrm, row, col, bufH, E);
  bias_act<true><<<blkNH, TB, 0, stream>>>(bufH, b1, NH);

  // ---- layer 2 ----
  gemm64_wmma<<<blkGemm, TB, 0, stream>>>(bufH, W2, bufT, N);
  zero_f32<<<2048, TB, 0, stream>>>(bufH, NH);
  scatter_kernel<<<blkScatter, TB, 0, stream>>>(bufT, nrm, row, col, bufH, E);
  bias_act<true><<<blkNH, TB, 0, stream>>>(bufH, b2, NH);

  // ---- layer 3 (no relu) ----
  gemm64_wmma<<<blkGemm, TB, 0, stream>>>(bufH, W3, bufT, N);
  zero_f32<<<2048, TB, 0, stream>>>(bufH, NH);
  scatter_kernel<<<blkScatter, TB, 0, stream>>>(bufT, nrm, row, col, bufH, E);
  bias_act<false><<<blkNH, TB, 0, stream>>>(bufH, b3, NH);

  // ---- pool + head ----
  zero_f32<<<64, TB, 0, stream>>>(gemb, (size_t)NG * HID);
  pool_kernel<<<blkNH, TB, 0, stream>>>(bufH, bat, gemb, NH);
  head_kernel<<<(NG + TB - 1) / TB, TB, 0, stream>>>(gemb, Wl, bl, out, NG);
}